// Uni_14456859918367
// MI455X (gfx1250) — compile-verified
//
#include <hip/hip_runtime.h>
#include <hip/hip_bf16.h>

typedef __attribute__((ext_vector_type(2))) float v2f;
typedef __attribute__((ext_vector_type(8))) float v8f;

#define DHID 64
#define TTAY 10

// ---------------- generic helpers ----------------

__global__ void zero_f(float* __restrict__ p, int n) {
    int g = blockIdx.x * blockDim.x + threadIdx.x;
    if (g < n) p[g] = 0.0f;
}

// deg[dst[e]] += w[e]
__global__ void scatter_deg(const int* __restrict__ dst, const float* __restrict__ w,
                            float* __restrict__ deg, int E) {
    int e = blockIdx.x * blockDim.x + threadIdx.x;
    if (e < E) atomicAdd(&deg[dst[e]], w[e]);
}

__global__ void make_dinv(const float* __restrict__ deg, float* __restrict__ dinv, int n) {
    int g = blockIdx.x * blockDim.x + threadIdx.x;
    if (g < n) {
        float d = deg[g];
        dinv[g] = (d > 0.0f) ? __frsqrt_rn(d) : 0.0f;
    }
}

__global__ void make_norm(const int* __restrict__ src, const int* __restrict__ dst,
                          const float* __restrict__ w, const float* __restrict__ dinv,
                          float* __restrict__ norm, int E) {
    int e = blockIdx.x * blockDim.x + threadIdx.x;
    if (e < E) norm[e] = dinv[src[e]] * w[e] * dinv[dst[e]];
}

// scalar propagate: out[dst[e]] += val[src[e]] * norm[e]
__global__ void scatter_scalar(const float* __restrict__ val, float* __restrict__ out,
                               const int* __restrict__ src, const int* __restrict__ dst,
                               const float* __restrict__ norm, int E) {
    int e = blockIdx.x * blockDim.x + threadIdx.x;
    if (e < E) atomicAdd(&out[dst[e]], val[src[e]] * norm[e]);
}

// h[n,d] = p[n] * W0[d] + b0[d]   (rank-1 expansion for block 0)
__global__ void expand_h(const float* __restrict__ p, const float* __restrict__ W0,
                         const float* __restrict__ b0, float* __restrict__ h, int n) {
    int g = blockIdx.x * blockDim.x + threadIdx.x;
    if (g < n * DHID) {
        int row = g >> 6;
        int d   = g & 63;
        h[g] = p[row] * W0[d] + b0[d];
    }
}

// ---------------- 64-wide propagate (blocks 1..10) ----------------
// one wave per edge; each lane handles 2 consecutive channels
__global__ void scatter64(const float* __restrict__ Hin, float* __restrict__ Hout,
                          const int* __restrict__ src, const int* __restrict__ dst,
                          const float* __restrict__ norm, int E) {
    unsigned g = blockIdx.x * blockDim.x + threadIdx.x;
    unsigned e = g >> 5;
    if (e >= (unsigned)E) return;
    int r = g & 31;
    int s = src[e], d = dst[e];
    float w = norm[e];
    float2 hv = *(const float2*)(Hin + (size_t)s * DHID + 2 * r);
    float* op = Hout + (size_t)d * DHID + 2 * r;
    atomicAdd(op + 0, hv.x * w);
    atomicAdd(op + 1, hv.y * w);
}

// ---------------- Q = expm(B - B^T) via 10-term Taylor, one block per matrix ----------------
__global__ void compute_q(const float* __restrict__ Wskew, float* __restrict__ Qout) {
    __shared__ float S[DHID * DHID];
    __shared__ float term[DHID * DHID];
    __shared__ float nterm[DHID * DHID];
    __shared__ float acc[DHID * DHID];
    const float* B = Wskew + (size_t)blockIdx.x * DHID * DHID;
    for (int i = threadIdx.x; i < DHID * DHID; i += blockDim.x) {
        int r = i >> 6, c = i & 63;
        S[i] = B[r * DHID + c] - B[c * DHID + r];
        float id = (r == c) ? 1.0f : 0.0f;
        term[i] = id;
        acc[i]  = id;
    }
    __syncthreads();
    for (int k = 1; k <= TTAY; ++k) {
        float invk = 1.0f / (float)k;
        for (int i = threadIdx.x; i < DHID * DHID; i += blockDim.x) {
            int r = i >> 6, c = i & 63;
            float sum = 0.0f;
            #pragma unroll 8
            for (int j = 0; j < DHID; ++j) sum += term[r * DHID + j] * S[j * DHID + c];
            nterm[i] = sum * invk;
        }
        __syncthreads();
        for (int i = threadIdx.x; i < DHID * DHID; i += blockDim.x) {
            term[i] = nterm[i];
            acc[i] += nterm[i];
        }
        __syncthreads();
    }
    for (int i = threadIdx.x; i < DHID * DHID; i += blockDim.x)
        Qout[(size_t)blockIdx.x * DHID * DHID + i] = acc[i];
}

// ---------------- Hout = Hin @ Q via V_WMMA_F32_16X16X4_F32 ----------------
// block = 256 threads = 8 waves; each wave computes one 16x64 row tile.
// A frag (16x4 f32): lane = m + 16*(k/2), reg = k%2  -> contiguous float2 per lane
// B frag (4x16 f32): lane = n + 16*(k/2), reg = k%2
// C/D (16x16 f32):   reg v holds M = v + 8*(lane/16), N = lane%16
__global__ void gemm_q_wmma(const float* __restrict__ Hin, const float* __restrict__ Q,
                            float* __restrict__ Hout, int nrows) {
    __shared__ float sQ[DHID * DHID];
    for (int i = threadIdx.x; i < DHID * DHID; i += blockDim.x) sQ[i] = Q[i];
    __syncthreads();

    int wave = threadIdx.x >> 5;
    int lane = threadIdx.x & 31;
    int tile = blockIdx.x * 8 + wave;            // 16-row tile index
    if (tile * 16 >= nrows) return;              // wave-uniform: EXEC stays all-1 inside

    int m       = lane & 15;
    int khalf   = lane >> 4;                     // kk offset = 2*khalf
    int row     = tile * 16 + m;
    int row_c   = (row < nrows) ? row : (nrows - 1);

    v8f c0 = {}, c1 = {}, c2 = {}, c3 = {};
    #pragma unroll
    for (int ks = 0; ks < 16; ++ks) {
        v2f a = *(const v2f*)(Hin + (size_t)row_c * DHID + ks * 4 + 2 * khalf);
        int krow = ks * 4 + 2 * khalf;
        {
            v2f b; b.x = sQ[(krow + 0) * DHID + 0 * 16 + m];
                   b.y = sQ[(krow + 1) * DHID + 0 * 16 + m];
            c0 = __builtin_amdgcn_wmma_f32_16x16x4_f32(false, a, false, b, (short)0, c0, false, false);
        }
        {
            v2f b; b.x = sQ[(krow + 0) * DHID + 1 * 16 + m];
                   b.y = sQ[(krow + 1) * DHID + 1 * 16 + m];
            c1 = __builtin_amdgcn_wmma_f32_16x16x4_f32(false, a, false, b, (short)0, c1, false, false);
        }
        {
            v2f b; b.x = sQ[(krow + 0) * DHID + 2 * 16 + m];
                   b.y = sQ[(krow + 1) * DHID + 2 * 16 + m];
            c2 = __builtin_amdgcn_wmma_f32_16x16x4_f32(false, a, false, b, (short)0, c2, false, false);
        }
        {
            v2f b; b.x = sQ[(krow + 0) * DHID + 3 * 16 + m];
                   b.y = sQ[(krow + 1) * DHID + 3 * 16 + m];
            c3 = __builtin_amdgcn_wmma_f32_16x16x4_f32(false, a, false, b, (short)0, c3, false, false);
        }
    }

    int m_base = tile * 16 + (lane >> 4) * 8;    // rows written by this lane: m_base + v
    int n_lane = lane & 15;
    #pragma unroll
    for (int v = 0; v < 8; ++v) {
        int orow = m_base + v;
        if (orow < nrows) {
            float* op = Hout + (size_t)orow * DHID + n_lane;
            op[0 * 16] = c0[v];
            op[1 * 16] = c1[v];
            op[2 * 16] = c2[v];
            op[3 * 16] = c3[v];
        }
    }
}

// ---------------- block 11 head: y[n] = dot(h[n,:], W11) ----------------
// one wave per row, lane handles 2 channels, shuffle reduce
__global__ void dot_w11(const float* __restrict__ H, const float* __restrict__ W11,
                        float* __restrict__ y, int n) {
    unsigned g = blockIdx.x * blockDim.x + threadIdx.x;
    unsigned row = g >> 5;
    int lane = g & 31;
    if (row >= (unsigned)n) return;
    float2 hv = *(const float2*)(H + (size_t)row * DHID + 2 * lane);
    float2 wv = *(const float2*)(W11 + 2 * lane);
    float p = hv.x * wv.x + hv.y * wv.y;
    #pragma unroll
    for (int off = 16; off > 0; off >>= 1) p += __shfl_xor(p, off, 32);
    if (lane == 0) y[row] = p;
}

__global__ void init_out(float* __restrict__ out, const float* __restrict__ b11, int n) {
    int g = blockIdx.x * blockDim.x + threadIdx.x;
    if (g < n) out[g] = b11[0];
}

// ---------------- launch ----------------

static inline unsigned cdiv(long long a, long long b) { return (unsigned)((a + b - 1) / b); }

extern "C" void kernel_launch(void* const* d_in, const int* in_sizes, int n_in,
                              void* d_out, int out_size, void* d_ws, size_t ws_size,
                              hipStream_t stream) {
    const float* x      = (const float*)d_in[0];   // [N,1]
    const int*   eidx   = (const int*)  d_in[1];   // [2,E]
    const float* ew     = (const float*)d_in[2];   // [E]
    const float* W0     = (const float*)d_in[3];   // [1,64]
    const float* b0     = (const float*)d_in[4];   // [64]
    const float* W11    = (const float*)d_in[5];   // [64,1]
    const float* b11    = (const float*)d_in[6];   // [1]
    const float* Wskew  = (const float*)d_in[7];   // [10,64,64]

    const int N    = in_sizes[0];
    const int E    = in_sizes[2];
    const int NMID = in_sizes[7] / (DHID * DHID);

    const int* src = eidx;
    const int* dst = eidx + E;

    // workspace layout (floats)
    float* bufA = (float*)d_ws;                    // [N,64]
    float* bufB = bufA + (size_t)N * DHID;         // [N,64]
    float* norm = bufB + (size_t)N * DHID;         // [E]
    float* deg  = norm + E;                        // [N]
    float* dinv = deg + N;                         // [N]
    float* pvec = dinv + N;                        // [N]
    float* yvec = pvec + N;                        // [N]
    float* Qbuf = yvec + N;                        // [NMID,64,64]
    float* outp = (float*)d_out;                   // [N]

    const int TB = 256;

    // --- GCN normalization ---
    zero_f<<<cdiv(N, TB), TB, 0, stream>>>(deg, N);
    scatter_deg<<<cdiv(E, TB), TB, 0, stream>>>(dst, ew, deg, E);
    make_dinv<<<cdiv(N, TB), TB, 0, stream>>>(deg, dinv, N);
    make_norm<<<cdiv(E, TB), TB, 0, stream>>>(src, dst, ew, dinv, norm, E);

    // --- block 0: h = propagate(x) (scalar) expanded by W0, + b0 ---
    zero_f<<<cdiv(N, TB), TB, 0, stream>>>(pvec, N);
    scatter_scalar<<<cdiv(E, TB), TB, 0, stream>>>(x, pvec, src, dst, norm, E);
    expand_h<<<cdiv((long long)N * DHID, TB), TB, 0, stream>>>(pvec, W0, b0, bufA, N);

    // --- orthogonal Q matrices (10x 64x64 Taylor expm) ---
    compute_q<<<NMID, TB, 0, stream>>>(Wskew, Qbuf);

    // --- blocks 1..10: h = propagate(h) @ Q_i  (ping-pong bufA <-> bufB) ---
    const unsigned gemm_blocks = cdiv((N + 15) / 16, 8);
    for (int i = 0; i < NMID; ++i) {
        zero_f<<<cdiv((long long)N * DHID, TB), TB, 0, stream>>>(bufB, N * DHID);
        scatter64<<<cdiv((long long)E * 32, TB), TB, 0, stream>>>(bufA, bufB, src, dst, norm, E);
        gemm_q_wmma<<<gemm_blocks, TB, 0, stream>>>(bufB, Qbuf + (size_t)i * DHID * DHID, bufA, N);
    }

    // --- block 11: y = h @ W11 (per-row dot), out = propagate(y) + b11 ---
    dot_w11<<<cdiv((long long)N * 32, TB), TB, 0, stream>>>(bufA, W11, yvec, N);
    init_out<<<cdiv(N, TB), TB, 0, stream>>>(outp, b11, N);
    scatter_scalar<<<cdiv(E, TB), TB, 0, stream>>>(yvec, outp, src, dst, norm, E);
}